// EdgeClassifierWrapper_53094385713662
// MI455X (gfx1250) — compile-verified
//
#include <hip/hip_runtime.h>
#include <hip/hip_bf16.h>

#define N_NODES 100000
#define N_EDGES 1600000
#define HID 128
#define IN_CH 6
#define BN_EPS 1e-5f

// packed-B layout constants: per 32-k slice, 16 k-pairs, pair-row stride 144 cols,
// 2 floats (k, k+1) interleaved per column -> 16*144*2 = 4608 floats per slice
#define BPAIR_STRIDE 144
#define BSLICE_ELEMS (16 * BPAIR_STRIDE * 2)  // 4608
#define APAD 36

typedef __attribute__((ext_vector_type(2))) float v2f;
typedef __attribute__((ext_vector_type(8))) float v8f;

__device__ __forceinline__ v8f wmma_f32(v2f a, v2f b, v8f c) {
  // V_WMMA_F32_16X16X4_F32 : D = A(16x4) x B(4x16) + C(16x16), all fp32
  return __builtin_amdgcn_wmma_f32_16x16x4_f32(false, a, false, b, (short)0, c,
                                               false, false);
}

// ---------------- degree / CSR construction ----------------

__global__ void k_init(int* cnt, int* cursor, int n) {
  int i = blockIdx.x * blockDim.x + threadIdx.x;
  if (i < n) { cnt[i] = 0; cursor[i] = 0; }
}

__global__ void k_count(const long long* ei, int* cnt, int E) {
  int e = blockIdx.x * blockDim.x + threadIdx.x;
  if (e < E) atomicAdd(&cnt[(int)ei[E + e]], 1);
}

__global__ void k_dinv(const int* cnt, float* dinv, int n) {
  int i = blockIdx.x * blockDim.x + threadIdx.x;
  if (i < n) dinv[i] = rsqrtf((float)cnt[i] + 1.0f);  // deg = in-count + self loop
}

__global__ void k_scanA(const int* cnt, int* offs, int* bsum, int n) {
  __shared__ int s[256];
  int i = blockIdx.x * 256 + threadIdx.x;
  int v = (i < n) ? cnt[i] : 0;
  s[threadIdx.x] = v;
  __syncthreads();
  for (int d = 1; d < 256; d <<= 1) {
    int t = (threadIdx.x >= (unsigned)d) ? s[threadIdx.x - d] : 0;
    __syncthreads();
    s[threadIdx.x] += t;
    __syncthreads();
  }
  if (i < n) offs[i] = s[threadIdx.x] - v;  // exclusive
  if (threadIdx.x == 255) bsum[blockIdx.x] = s[255];
}

__global__ void k_scanB(int* bsum, int* offs, int nb, int n) {
  if (blockIdx.x == 0 && threadIdx.x == 0) {
    int acc = 0;
    for (int b = 0; b < nb; b++) { int t = bsum[b]; bsum[b] = acc; acc += t; }
    offs[n] = acc;  // == E
  }
}

__global__ void k_scanC(int* offs, const int* bsum, int n) {
  int i = blockIdx.x * 256 + threadIdx.x;
  if (i < n) offs[i] += bsum[blockIdx.x];
}

__global__ void k_fill(const long long* ei, const int* offs, int* cursor,
                       int* csr_src, int E) {
  int e = blockIdx.x * blockDim.x + threadIdx.x;
  if (e < E) {
    int s = (int)ei[e];
    int d = (int)ei[E + e];
    int pos = offs[d] + atomicAdd(&cursor[d], 1);
    csr_src[pos] = s;
  }
}

// Sort each node's segment by src id: makes FP accumulation order deterministic.
__global__ void k_sortseg(const int* offs, int* csr_src, int n) {
  int i = blockIdx.x * blockDim.x + threadIdx.x;
  if (i >= n) return;
  int a = offs[i], b = offs[i + 1];
  for (int j = a + 1; j < b; j++) {
    int key = csr_src[j];
    int k = j - 1;
    while (k >= a && csr_src[k] > key) { csr_src[k + 1] = csr_src[k]; k--; }
    csr_src[k + 1] = key;
  }
}

// ---------------- weight pre-pack into WMMA-friendly interleaved layout ----------------
// Bp[slice][pair=(k&31)>>1][n][k&1] with pair-row stride BPAIR_STRIDE.

__global__ void k_packB(const float* __restrict__ W, float* __restrict__ Bp,
                        int rows) {
  int idx = blockIdx.x * 256 + threadIdx.x;
  if (idx >= rows * HID) return;
  int k = idx >> 7, n = idx & 127;
  int s = k >> 5;
  Bp[(size_t)s * BSLICE_ELEMS + ((k & 31) >> 1) * (BPAIR_STRIDE * 2) + (n << 1) +
     (k & 1)] = W[idx];
}

// ---------------- layer-0 linear (K=6, VALU) ----------------

__global__ void k_lin0(const float* __restrict__ x, const float* __restrict__ W0,
                       float* __restrict__ h0) {
  int n = blockIdx.x;
  int c = threadIdx.x;
  float acc = 0.f;
#pragma unroll
  for (int k = 0; k < IN_CH; k++) acc += x[n * IN_CH + k] * W0[k * HID + c];
  h0[n * HID + c] = acc;
}

// ------------- gather aggregation + bias + BN + ReLU -------------

__global__ void k_agg(const float* __restrict__ hin, const float* __restrict__ dinv,
                      const int* __restrict__ offs, const int* __restrict__ csr_src,
                      const float* __restrict__ b, const float* __restrict__ g,
                      const float* __restrict__ be, const float* __restrict__ m,
                      const float* __restrict__ v, float* __restrict__ hout) {
  int n = blockIdx.x;
  int c = threadIdx.x;
  float dn = dinv[n];
  float acc = hin[(size_t)n * HID + c] * dn;  // self loop (dst factor applied at end)
  int j0 = offs[n], j1 = offs[n + 1];
  for (int j = j0; j < j1; j++) {
    int s = csr_src[j];
    acc += hin[(size_t)s * HID + c] * dinv[s];
  }
  float val = acc * dn + b[c];
  val = (val - m[c]) * rsqrtf(v[c] + BN_EPS) * g[c] + be[c];
  hout[(size_t)n * HID + c] = fmaxf(val, 0.f);
}

// ---------------- WMMA GEMM: t = h1 @ W1 (N x 128 x 128), W1 pre-packed ----------------

__global__ __launch_bounds__(256) void k_gemm(const float* __restrict__ A,
                                              const float* __restrict__ Bp,
                                              float* __restrict__ C, int nrows) {
  __shared__ float As[128 * APAD];
  __shared__ float Bs[BSLICE_ELEMS];
  int tid = threadIdx.x;
  int wave = tid >> 5, lane = tid & 31;
  int half = lane >> 4, lm = lane & 15;
  int rowBase = blockIdx.x * 128;

  v8f zero = {0.f, 0.f, 0.f, 0.f, 0.f, 0.f, 0.f, 0.f};
  v8f acc[8];
#pragma unroll
  for (int t = 0; t < 8; t++) acc[t] = zero;

  for (int s = 0; s < 4; s++) {
    int ks = s * 32;
    // stage A slice: 128 rows x 32 cols
#pragma unroll
    for (int i = 0; i < 4; i++) {
      int chunk = tid + i * 256;
      int r = chunk >> 3, c4 = (chunk & 7) * 4;
      int gr = min(rowBase + r, nrows - 1);
      float4 val = *(const float4*)(A + (size_t)gr * HID + ks + c4);
      *(float4*)&As[r * APAD + c4] = val;
    }
    // stage packed B slice: straight contiguous copy (1152 float4)
#pragma unroll
    for (int i = 0; i < 5; i++) {
      int chunk = tid + i * 256;
      if (chunk < BSLICE_ELEMS / 4) {
        float4 val = *(const float4*)(Bp + (size_t)s * BSLICE_ELEMS + chunk * 4);
        *(float4*)&Bs[chunk * 4] = val;
      }
    }
    __syncthreads();
#pragma unroll
    for (int kk = 0; kk < 32; kk += 4) {
      v2f a = *(const v2f*)&As[(wave * 16 + lm) * APAD + kk + 2 * half];
      int pb = ((kk >> 1) + half) * (BPAIR_STRIDE * 2) + lm * 2;
      v2f bf[8];
#pragma unroll
      for (int t = 0; t < 8; t++) bf[t] = *(const v2f*)&Bs[pb + t * 32];
#pragma unroll
      for (int t = 0; t < 8; t++) acc[t] = wmma_f32(a, bf[t], acc[t]);
    }
    __syncthreads();
  }
#pragma unroll
  for (int t = 0; t < 8; t++) {
    int n = t * 16 + lm;
#pragma unroll
    for (int v = 0; v < 8; v++) {
      int r = rowBase + wave * 16 + v + 8 * half;
      if (r < nrows) C[(size_t)r * HID + n] = acc[t][v];
    }
  }
}

// -------- edge MLP: logits = relu([h_src|h_dst|ea] @ Wc1 + bc1) @ Wc2 + bc2 --------
// Wc1 rows 0..255 pre-packed; tail rows 256,257 applied via VALU.

__global__ __launch_bounds__(256) void k_edge_mlp(
    const float* __restrict__ h, const long long* __restrict__ ei,
    const float* __restrict__ ea, const float* __restrict__ Bp,
    const float* __restrict__ Wc1, const float* __restrict__ bc1,
    const float* __restrict__ Wc2, const float* __restrict__ bc2,
    float* __restrict__ out, int E) {
  __shared__ float As[128 * APAD];
  __shared__ float Bs[BSLICE_ELEMS];
  __shared__ int snode[2][128];
  __shared__ float sea[128][2];
  __shared__ float stail[2][128];
  __shared__ float sbias[128];
  __shared__ float swc2[128];

  int tid = threadIdx.x;
  int wave = tid >> 5, lane = tid & 31;
  int half = lane >> 4, lm = lane & 15;
  long long eBase = (long long)blockIdx.x * 128;

  if (tid < 128) {
    long long e = eBase + tid;
    if (e > (long long)E - 1) e = (long long)E - 1;
    snode[0][tid] = (int)ei[e];
    snode[1][tid] = (int)ei[(long long)E + e];
    sea[tid][0] = ea[e * 2 + 0];
    sea[tid][1] = ea[e * 2 + 1];
    sbias[tid] = bc1[tid];
    swc2[tid] = Wc2[tid];
  } else {
    int c = tid - 128;
    stail[0][c] = Wc1[256 * HID + c];
    stail[1][c] = Wc1[257 * HID + c];
  }
  __syncthreads();

  v8f zero = {0.f, 0.f, 0.f, 0.f, 0.f, 0.f, 0.f, 0.f};
  v8f acc[8];
#pragma unroll
  for (int t = 0; t < 8; t++) acc[t] = zero;

  for (int s = 0; s < 8; s++) {            // K = 256 in 8 slices of 32
    int tbl = s >> 2;                      // 0: h[src] cols, 1: h[dst] cols
    int colBase = (s & 3) * 32;
#pragma unroll
    for (int i = 0; i < 4; i++) {
      int chunk = tid + i * 256;
      int r = chunk >> 3, c4 = (chunk & 7) * 4;
      int node = snode[tbl][r];
      float4 val = *(const float4*)(h + (size_t)node * HID + colBase + c4);
      *(float4*)&As[r * APAD + c4] = val;
    }
#pragma unroll
    for (int i = 0; i < 5; i++) {
      int chunk = tid + i * 256;
      if (chunk < BSLICE_ELEMS / 4) {
        float4 val = *(const float4*)(Bp + (size_t)s * BSLICE_ELEMS + chunk * 4);
        *(float4*)&Bs[chunk * 4] = val;
      }
    }
    __syncthreads();
#pragma unroll
    for (int kk = 0; kk < 32; kk += 4) {
      v2f a = *(const v2f*)&As[(wave * 16 + lm) * APAD + kk + 2 * half];
      int pb = ((kk >> 1) + half) * (BPAIR_STRIDE * 2) + lm * 2;
      v2f bf[8];
#pragma unroll
      for (int t = 0; t < 8; t++) bf[t] = *(const v2f*)&Bs[pb + t * 32];
#pragma unroll
      for (int t = 0; t < 8; t++) acc[t] = wmma_f32(a, bf[t], acc[t]);
    }
    __syncthreads();
  }

  // edge_attr tail + bias + ReLU, then dot with Wc2
  float partial[8];
#pragma unroll
  for (int v = 0; v < 8; v++) partial[v] = 0.f;
#pragma unroll
  for (int t = 0; t < 8; t++) {
    int n = t * 16 + lm;
    float w0 = stail[0][n], w1 = stail[1][n];
    float bias = sbias[n], w2 = swc2[n];
#pragma unroll
    for (int v = 0; v < 8; v++) {
      int r = wave * 16 + v + 8 * half;
      float z = acc[t][v] + bias + sea[r][0] * w0 + sea[r][1] * w1;
      z = fmaxf(z, 0.f);
      partial[v] += z * w2;
    }
  }
#pragma unroll
  for (int v = 0; v < 8; v++) {
    float p = partial[v];
    p += __shfl_xor(p, 1, 32);
    p += __shfl_xor(p, 2, 32);
    p += __shfl_xor(p, 4, 32);
    p += __shfl_xor(p, 8, 32);
    partial[v] = p;
  }
  if (lm == 0) {  // lanes 0 (rows 0-7) and 16 (rows 8-15)
    float b2 = bc2[0];
#pragma unroll
    for (int v = 0; v < 8; v++) {
      long long e = eBase + wave * 16 + v + 8 * half;
      if (e < (long long)E) out[e] = partial[v] + b2;
    }
  }
}

// ---------------- host launcher ----------------

extern "C" void kernel_launch(void* const* d_in, const int* in_sizes, int n_in,
                              void* d_out, int out_size, void* d_ws, size_t ws_size,
                              hipStream_t stream) {
  (void)in_sizes; (void)n_in; (void)out_size; (void)ws_size;
  const float* x = (const float*)d_in[0];
  const long long* ei = (const long long*)d_in[1];
  const float* ea = (const float*)d_in[2];
  const float* W0 = (const float*)d_in[3];
  const float* b0 = (const float*)d_in[4];
  const float* g0 = (const float*)d_in[5];
  const float* be0 = (const float*)d_in[6];
  const float* m0 = (const float*)d_in[7];
  const float* v0 = (const float*)d_in[8];
  const float* W1 = (const float*)d_in[9];
  const float* b1 = (const float*)d_in[10];
  const float* g1 = (const float*)d_in[11];
  const float* be1 = (const float*)d_in[12];
  const float* m1 = (const float*)d_in[13];
  const float* v1 = (const float*)d_in[14];
  const float* Wc1 = (const float*)d_in[15];
  const float* bc1 = (const float*)d_in[16];
  const float* Wc2 = (const float*)d_in[17];
  const float* bc2 = (const float*)d_in[18];
  float* out = (float*)d_out;

  const int N = N_NODES, E = N_EDGES;
  char* p = (char*)d_ws;
  float* bufA = (float*)p; p += (size_t)N * HID * sizeof(float);   // h0, then t
  float* bufB = (float*)p; p += (size_t)N * HID * sizeof(float);   // h1, then h2
  int* cnt = (int*)p;      p += (size_t)N * sizeof(int);
  int* cursor = (int*)p;   p += (size_t)N * sizeof(int);
  float* dinv = (float*)p; p += (size_t)N * sizeof(float);
  int* offs = (int*)p;     p += ((size_t)N + 64) * sizeof(int);
  int* bsum = (int*)p;     p += 1024 * sizeof(int);
  int* csr_src = (int*)p;  p += (size_t)E * sizeof(int);
  float* BpW1 = (float*)p; p += (size_t)4 * BSLICE_ELEMS * sizeof(float);
  float* BpWc1 = (float*)p; p += (size_t)8 * BSLICE_ELEMS * sizeof(float);

  int nbN = (N + 255) / 256;
  int nbE = (E + 255) / 256;

  k_init<<<nbN, 256, 0, stream>>>(cnt, cursor, N);
  k_count<<<nbE, 256, 0, stream>>>(ei, cnt, E);
  k_dinv<<<nbN, 256, 0, stream>>>(cnt, dinv, N);
  k_scanA<<<nbN, 256, 0, stream>>>(cnt, offs, bsum, N);
  k_scanB<<<1, 32, 0, stream>>>(bsum, offs, nbN, N);
  k_scanC<<<nbN, 256, 0, stream>>>(offs, bsum, N);
  k_fill<<<nbE, 256, 0, stream>>>(ei, offs, cursor, csr_src, E);
  k_sortseg<<<nbN, 256, 0, stream>>>(offs, csr_src, N);

  // pre-pack weights for WMMA-friendly LDS staging
  k_packB<<<(128 * HID + 255) / 256, 256, 0, stream>>>(W1, BpW1, 128);
  k_packB<<<(256 * HID + 255) / 256, 256, 0, stream>>>(Wc1, BpWc1, 256);

  // Layer 0: h0 = x @ W0 ; h1 = relu(bn(agg(h0) + b0))
  k_lin0<<<N, HID, 0, stream>>>(x, W0, bufA);
  k_agg<<<N, HID, 0, stream>>>(bufA, dinv, offs, csr_src, b0, g0, be0, m0, v0, bufB);

  // Layer 1: t = h1 @ W1 ; h2 = relu(bn(agg(t) + b1))
  int gemmBlocks = (N + 127) / 128;
  k_gemm<<<gemmBlocks, 256, 0, stream>>>(bufB, BpW1, bufA, N);
  k_agg<<<N, HID, 0, stream>>>(bufA, dinv, offs, csr_src, b1, g1, be1, m1, v1, bufB);

  // Edge classifier
  int mlpBlocks = (E + 127) / 128;
  k_edge_mlp<<<mlpBlocks, 256, 0, stream>>>(bufB, ei, ea, BpWc1, Wc1, bc1, Wc2, bc2,
                                            out, E);
}